// Interp23Tap_41781441855986
// MI455X (gfx1250) — compile-verified
//
#include <hip/hip_runtime.h>
#include <hip/hip_bf16.h>

typedef __attribute__((ext_vector_type(2))) float v2f;
typedef __attribute__((ext_vector_type(8))) float v8f;

#define TAPS 23

// ---------------------------------------------------------------------------
// Build the composed per-dimension operator W[64][48] on device.
// Out 64-row tile (origin 64R) = sum_{a=-1..1} W_a * In 16-row block (R+a);
// W[u][j] maps input row j of the 48-row window (origin 16(R-1)) to output
// row u of the 64-row tile (origin 64R -> local y = 64+u).
//   stage 1 (off=1): up1[2i+1] = x[i];  f1[m] = sum_s k[s] up1[m+s-11]
//   stage 2 (off=0): up2[2m]  = f1[m];  f2[y] = sum_t k[t] up2[y+t-11]
// ---------------------------------------------------------------------------
__global__ void build_w_kernel(const float* __restrict__ ker,
                               float* __restrict__ Wmat) {
    const int u = threadIdx.x;           // 0..63
    if (u >= 64) return;
    const int yloc = 64 + u;             // local coord, origin 64*(R-1)
    for (int j = 0; j < 48; ++j) {
        float acc = 0.0f;
        for (int t = 0; t < TAPS; ++t) {
            const int p = yloc + t - 11;       // up2 index (stage-2 off = 0)
            if (p & 1) continue;
            const int m = p >> 1;              // f1 index
            const int s = 2 * j + 1 - m + 11;  // up1 index must be 2j+1 (off = 1)
            if (s >= 0 && s < TAPS) acc += ker[t] * ker[s];
        }
        Wmat[u * 48 + j] = acc;
    }
}

// ---------------------------------------------------------------------------
// One workgroup = one 64x64 output tile of one (b,c) slice.
// 128 threads = 4 waves; wave w produces output rows [16w, 16w+16) of tile.
//   Step 1: T[64x48] = W[64x48] * In[48x48]   (vertical operator)
//   Step 2: Out[64x64] = T * W^T              (horizontal operator)
// All products via v_wmma_f32_16x16x4_f32, K accumulated in chains of 12.
// ---------------------------------------------------------------------------
__launch_bounds__(128)
__global__ void upsample4x_wmma_kernel(const float* __restrict__ x,
                                       const float* __restrict__ Wmat,
                                       float* __restrict__ out) {
    __shared__ float sW[64 * 48];        // 12 KB  operator
    __shared__ float sIn[48 * 48];       //  9 KB  input window
    __shared__ float sT[4][16 * 48];     // 12 KB  per-wave step-1 result

    const int tid   = threadIdx.x;
    const int wave  = tid >> 5;          // 0..3  (= mt, 16-row strip of tile)
    const int lane  = tid & 31;
    const int bid   = blockIdx.x;
    const int slice = bid >> 8;          // 0..31  (b*8+c)
    const int R     = (bid >> 4) & 15;   // tile row   (64-px units)
    const int C     = bid & 15;          // tile col   (64-px units)

    // Stage operator into LDS (3072 floats, 24 per thread).
    for (int i = tid; i < 64 * 48; i += 128) sW[i] = Wmat[i];

    // Stage 48x48 input window (3x3 blocks of 16, circular block wrap).
    const float* xs = x + (size_t)slice * 256 * 256;
    for (int i = tid; i < 48 * 48; i += 128) {
        const int r  = i / 48;
        const int c  = i - r * 48;
        const int gr = (R * 16 + 240 + r) & 255;   // +240 == -16 mod 256
        const int gc = (C * 16 + 240 + c) & 255;
        sIn[i] = xs[gr * 256 + gc];
    }
    __syncthreads();

    const int mrow  = lane & 15;           // M for A / N for B / N for D
    const int hi    = lane >> 4;           // lane half
    const int khalf = hi << 1;             // K offset 0 or 2 within group of 4
    const int mt    = wave;

    // ---- Step 1: T[mt] = W[mt-strip] x In --------------------------------
    for (int nt = 0; nt < 3; ++nt) {
        v8f acc = {};
        #pragma unroll
        for (int kb = 0; kb < 12; ++kb) {
            const int k0 = kb * 4 + khalf;
            v2f a, b;
            a.x = sW[(mt * 16 + mrow) * 48 + k0];
            a.y = sW[(mt * 16 + mrow) * 48 + k0 + 1];
            b.x = sIn[(k0)     * 48 + nt * 16 + mrow];
            b.y = sIn[(k0 + 1) * 48 + nt * 16 + mrow];
            acc = __builtin_amdgcn_wmma_f32_16x16x4_f32(
                false, a, false, b, (short)0, acc, false, false);
        }
        #pragma unroll
        for (int r8 = 0; r8 < 8; ++r8) {
            const int m = r8 + (hi << 3);            // D row
            sT[wave][m * 48 + nt * 16 + mrow] = acc[r8];
        }
    }
    __syncthreads();

    // ---- Step 2: Out[mt][nt] = T[mt] x W^T -------------------------------
    float* outs = out + (size_t)slice * 1024 * 1024;
    for (int nt = 0; nt < 4; ++nt) {
        v8f acc = {};
        #pragma unroll
        for (int kb = 0; kb < 12; ++kb) {
            const int k0 = kb * 4 + khalf;
            v2f a, b;
            a.x = sT[wave][mrow * 48 + k0];
            a.y = sT[wave][mrow * 48 + k0 + 1];
            b.x = sW[(nt * 16 + mrow) * 48 + k0];      // W^T(k,n) = W(n,k)
            b.y = sW[(nt * 16 + mrow) * 48 + k0 + 1];
            acc = __builtin_amdgcn_wmma_f32_16x16x4_f32(
                false, a, false, b, (short)0, acc, false, false);
        }
        #pragma unroll
        for (int r8 = 0; r8 < 8; ++r8) {
            const int m    = r8 + (hi << 3);
            const int grow = R * 64 + mt * 16 + m;
            const int gcol = C * 64 + nt * 16 + mrow;
            outs[(size_t)grow * 1024 + gcol] = acc[r8];
        }
    }
}

// ---------------------------------------------------------------------------
extern "C" void kernel_launch(void* const* d_in, const int* in_sizes, int n_in,
                              void* d_out, int out_size, void* d_ws, size_t ws_size,
                              hipStream_t stream) {
    const float* x   = (const float*)d_in[0];   // (4,8,256,256) f32
    const float* ker = (const float*)d_in[1];   // 23 taps f32
    float* Wmat = (float*)d_ws;                 // 64*48 f32 = 12 KB scratch

    build_w_kernel<<<1, 64, 0, stream>>>(ker, Wmat);
    upsample4x_wmma_kernel<<<32 * 16 * 16, 128, 0, stream>>>(
        x, Wmat, (float*)d_out);
}